// NoisyQuantumLayer_31095563223605
// MI455X (gfx1250) — compile-verified
//
#include <hip/hip_runtime.h>
#include <math.h>

#define NQ     6
#define DIM    64
#define DEPTH  4
#define BATCH  64
#define NELEM  (DIM * DIM)     // 4096
#define NBLK   1024            // 32x32 2x2-blocks per qubit
#define TPB    256

typedef float v2f __attribute__((ext_vector_type(2)));
typedef float v8f __attribute__((ext_vector_type(8)));

// insert a 0 bit at position p into 5-bit value u
__device__ __forceinline__ int expand_bit(int u, int p) {
    int low = u & ((1 << p) - 1);
    return ((u >> p) << (p + 1)) | low;
}

// A -> U^dag A U with U = RY(t) = [[c,-s],[s,c]] on bit position p (real coeffs)
__device__ __forceinline__ void conj_ry(float* Ar, float* Ai, int p, float c, float s, int tid) {
    for (int b = tid; b < NBLK; b += TPB) {
        int r0 = expand_bit(b >> 5, p), c0 = expand_bit(b & 31, p);
        int r1 = r0 | (1 << p), c1 = c0 | (1 << p);
        int i00 = r0 * DIM + c0, i01 = r0 * DIM + c1;
        int i10 = r1 * DIM + c0, i11 = r1 * DIM + c1;
        {
            float b00 = Ar[i00], b01 = Ar[i01], b10 = Ar[i10], b11 = Ar[i11];
            float t00 =  c * b00 + s * b10, t01 =  c * b01 + s * b11;
            float t10 = -s * b00 + c * b10, t11 = -s * b01 + c * b11;
            Ar[i00] =  t00 * c + t01 * s;  Ar[i01] = -t00 * s + t01 * c;
            Ar[i10] =  t10 * c + t11 * s;  Ar[i11] = -t10 * s + t11 * c;
        }
        {
            float b00 = Ai[i00], b01 = Ai[i01], b10 = Ai[i10], b11 = Ai[i11];
            float t00 =  c * b00 + s * b10, t01 =  c * b01 + s * b11;
            float t10 = -s * b00 + c * b10, t11 = -s * b01 + c * b11;
            Ai[i00] =  t00 * c + t01 * s;  Ai[i01] = -t00 * s + t01 * c;
            Ai[i10] =  t10 * c + t11 * s;  Ai[i11] = -t10 * s + t11 * c;
        }
    }
    __syncthreads();
}

// A -> RZ(t)^dag A RZ(t): A01 *= e^{+it}, A10 *= e^{-it} on bit position p
__device__ __forceinline__ void conj_rz(float* Ar, float* Ai, int p, float t, int tid) {
    float ct = cosf(t), st = sinf(t);
    for (int e = tid; e < NELEM; e += TPB) {
        int r = e >> 6, c = e & 63;
        int br = (r >> p) & 1, bc = (c >> p) & 1;
        if (br != bc) {
            float sgn = br ? -st : st;
            float re = Ar[e], im = Ai[e];
            Ar[e] = re * ct - im * sgn;
            Ai[e] = re * sgn + im * ct;
        }
    }
    __syncthreads();
}

// fused adjoint of (amp-damp ga then phase-damp) on bit position p:
// off-diag *= soff = sqrt(1-ga)*sqrt(1-gp);  a11 = (1-ga)*a11 + ga*a00
__device__ __forceinline__ void channel_adj(float* Ar, float* Ai, int p, float ga, float soff, int tid) {
    float om = 1.0f - ga;
    for (int b = tid; b < NBLK; b += TPB) {
        int r0 = expand_bit(b >> 5, p), c0 = expand_bit(b & 31, p);
        int r1 = r0 | (1 << p), c1 = c0 | (1 << p);
        int i00 = r0 * DIM + c0, i01 = r0 * DIM + c1;
        int i10 = r1 * DIM + c0, i11 = r1 * DIM + c1;
        Ar[i01] *= soff; Ai[i01] *= soff;
        Ar[i10] *= soff; Ai[i10] *= soff;
        Ar[i11] = om * Ar[i11] + ga * Ar[i00];
        Ai[i11] = om * Ai[i11] + ga * Ai[i00];
    }
    __syncthreads();
}

// A -> CNOT A CNOT (involutive permutation P on rows and cols)
__device__ __forceinline__ void cnot_conj(float* Ar, float* Ai, int pc, int pt, int tid) {
    for (int e = tid; e < NELEM; e += TPB) {
        int r = e >> 6, c = e & 63;
        int Pr = ((r >> pc) & 1) ? (r ^ (1 << pt)) : r;
        int Pc = ((c >> pc) & 1) ? (c ^ (1 << pt)) : c;
        int e2 = Pr * DIM + Pc;
        if (e < e2) {
            float tr = Ar[e]; Ar[e] = Ar[e2]; Ar[e2] = tr;
            float ti = Ai[e]; Ai[e] = Ai[e2]; Ai[e2] = ti;
        }
    }
    __syncthreads();
}

// ---------------- Kernel 1: back-propagate Z0 through the adjoint circuit ----------------
__global__ void __launch_bounds__(TPB)
nq_build_B_kernel(const float* __restrict__ w, float* __restrict__ ReB) {
    __shared__ float Ar[NELEM];
    __shared__ float Ai[NELEM];
    int tid = threadIdx.x;

    // A = Z0 (qubit 0 = MSB of the 6-bit index)
    for (int e = tid; e < NELEM; e += TPB) {
        int r = e >> 6, c = e & 63;
        Ar[e] = (r == c) ? ((r < 32) ? 1.0f : -1.0f) : 0.0f;
        Ai[e] = 0.0f;
    }
    __syncthreads();

    const float AD1 = 9e-5f,    PD1 = 6e-5f;
    const float AD2 = 1.95e-3f, PD2 = 1.3e-3f;
    const float soff1 = sqrtf(1.0f - AD1) * sqrtf(1.0f - PD1);
    const float soff2 = sqrtf(1.0f - AD2) * sqrtf(1.0f - PD2);

    for (int layer = DEPTH - 1; layer >= 0; --layer) {
        // reverse of: RY, RZ, AD1, PD1 on each qubit
        for (int i = NQ - 1; i >= 0; --i) {
            int p = 5 - i;
            channel_adj(Ar, Ai, p, AD1, soff1, tid);                 // PD1^ then AD1^ (fused)
            conj_rz(Ar, Ai, p, w[(layer * NQ + i) * 2 + 1], tid);    // RZ^dag A RZ
            float t0 = 0.5f * w[(layer * NQ + i) * 2 + 0];
            conj_ry(Ar, Ai, p, cosf(t0), sinf(t0), tid);             // RY^dag A RY
        }
        // reverse of: CNOT(i,i+1) then 2q noise on i, i+1
        for (int i = NQ - 2; i >= 0; --i) {
            int pc = 5 - i, pt = 4 - i;
            channel_adj(Ar, Ai, pt, AD2, soff2, tid);                // qubit i+1
            channel_adj(Ar, Ai, pc, AD2, soff2, tid);                // qubit i
            cnot_conj(Ar, Ai, pc, pt, tid);
        }
    }
    // encoding-layer noise adjoints (RY(x) stays in per-batch psi)
    for (int i = NQ - 1; i >= 0; --i)
        channel_adj(Ar, Ai, 5 - i, AD1, soff1, tid);

    for (int e = tid; e < NELEM; e += TPB)
        ReB[e] = Ar[e];   // B Hermitian; only Re(B) contributes to psi^T B psi
}

// ---------------- Kernel 2: out_b = psi_b^T * ReB * psi_b via fp32 WMMA ----------------
__global__ void __launch_bounds__(TPB)
nq_batch_expect_kernel(const float* __restrict__ x, const float* __restrict__ ReB,
                       float* __restrict__ out) {
    __shared__ float Bs[NELEM];
    __shared__ float Psi[NELEM];   // Psi[b][j]
    __shared__ float T[NELEM];     // T = Psi * Bs
    int tid = threadIdx.x;

    for (int e = tid; e < NELEM; e += TPB) Bs[e] = ReB[e];

    // product-state amplitudes: psi_b[j] = prod_i (bit ? sin(x/2) : cos(x/2)), qubit0 = MSB
    for (int e = tid; e < NELEM; e += TPB) {
        int b = e >> 6, j = e & 63;
        float prod = 1.0f;
#pragma unroll
        for (int i = 0; i < NQ; ++i) {
            float h = 0.5f * x[b * NQ + i];
            prod *= (((j >> (5 - i)) & 1) ? sinf(h) : cosf(h));
        }
        Psi[e] = prod;
    }
    __syncthreads();

    // 64x64x64 fp32 GEMM: 4x4 grid of 16x16 tiles; 8 waves x 2 tiles
    int wave = tid >> 5, lane = tid & 31;
    int ln = lane & 15;            // A's M / B's N sub-index
    int hi = lane >> 4;            // half-select
    int kk = hi * 2;               // K pair base for this lane
#pragma unroll
    for (int r = 0; r < 2; ++r) {
        int t  = wave * 2 + r;
        int tm = (t >> 2) * 16;    // tile row base (batch)
        int tn = (t & 3) * 16;     // tile col base
        v8f acc = {};
        for (int k0 = 0; k0 < DIM; k0 += 4) {
            v2f a, b;
            a.x = Psi[(tm + ln) * DIM + k0 + kk];
            a.y = Psi[(tm + ln) * DIM + k0 + kk + 1];
            b.x = Bs[(k0 + kk) * DIM + tn + ln];
            b.y = Bs[(k0 + kk + 1) * DIM + tn + ln];
            acc = __builtin_amdgcn_wmma_f32_16x16x4_f32(
                false, a, false, b, (short)0, acc, false, false);
        }
#pragma unroll
        for (int v = 0; v < 8; ++v) {
            int M = tm + v + 8 * hi;
            int N = tn + ln;
            T[M * DIM + N] = acc[v];
        }
    }
    __syncthreads();

    // out_b = sum_j T[b][j] * Psi[b][j]
    if (tid < BATCH) {
        float s = 0.0f;
        for (int j = 0; j < DIM; ++j)
            s += T[tid * DIM + j] * Psi[tid * DIM + j];
        out[tid] = s;
    }
}

extern "C" void kernel_launch(void* const* d_in, const int* in_sizes, int n_in,
                              void* d_out, int out_size, void* d_ws, size_t ws_size,
                              hipStream_t stream) {
    const float* x = (const float*)d_in[0];        // [64, 6] f32
    const float* w = (const float*)d_in[1];        // [4, 6, 2] f32
    float* out = (float*)d_out;                    // [64, 1] f32
    float* ReB = (float*)d_ws;                     // 4096 f32 scratch

    nq_build_B_kernel<<<1, TPB, 0, stream>>>(w, ReB);
    nq_batch_expect_kernel<<<1, TPB, 0, stream>>>(x, ReB, out);
}